// HybridTwoTowerModel_18906446037173
// MI455X (gfx1250) — compile-verified
//
#include <hip/hip_runtime.h>
#include <hip/hip_bf16.h>
#include <stdint.h>

#define N_NODES 100000
#define N_EDGES 1000000
#define BATCH   1024
#define RELS    8
#define HID     64
#define NGRAMD  4096
#define EMBD    16
#define OUTD    128
#define EPSV    1e-5f
#define KC      64          // K-chunk staged in LDS per workgroup
#define USE_TDM 1           // stage B panel via Tensor Data Mover

typedef __attribute__((ext_vector_type(2))) float    v2f;
typedef __attribute__((ext_vector_type(8))) float    v8f;
typedef __attribute__((ext_vector_type(4))) unsigned uint32x4;
typedef __attribute__((ext_vector_type(8))) int      int32x8;
typedef __attribute__((ext_vector_type(4))) int      int32x4;

__device__ __forceinline__ float act_apply(float v, int act) {
  if (act == 1) return v > 0.0f ? v : 0.2f * v;   // LeakyReLU(0.2)
  if (act == 2) return v > 0.0f ? v : 0.0f;       // ReLU
  return v;
}

// ---------------------------------------------------------------------------
// f32 WMMA GEMM with LDS-staged B panel (loaded by the Tensor Data Mover):
//   C[M,Nc](ldc) = act( diag(1/max(rowscale,1)) * A[M,K](lda) @ W[K,Nc] + bias [+ C] )
// flags: bit0 = accumulate into C, bits[2:1] = activation (0 none, 1 lrelu, 2 relu)
// Block = 8 wave32; block owns 8 row-tiles x 1 col-tile; B chunk (KC x 16) lives
// in LDS, DMA'd once per chunk (8x less B traffic than per-wave global loads).
// ---------------------------------------------------------------------------
__global__ void __launch_bounds__(256)
wmma_gemm_kernel(const float* __restrict__ A,
                 const float* __restrict__ W,
                 const float* __restrict__ bias,
                 const float* __restrict__ rowscale,
                 float* __restrict__ C,
                 int M, int K, int Nc, int lda, int ldc, int flags)
{
  __shared__ float ldsB[KC * 16];                  // 4 KB of the 320 KB WGP LDS

  const int lane     = threadIdx.x & 31;
  const int waveId   = threadIdx.x >> 5;
  const int colTiles = Nc >> 4;
  const int rowTiles = M >> 4;
  const int colTile  = blockIdx.x % colTiles;
  const int rowGroup = blockIdx.x / colTiles;
  const int rowTile  = rowGroup * 8 + waveId;
  const bool valid   = (rowTile < rowTiles);       // wave-uniform

  const int half = lane >> 4;                      // 0: lanes 0-15, 1: lanes 16-31
  const int l    = lane & 15;
  const int arow = (valid ? rowTile : (rowTiles - 1)) * 16 + l;
  const int bcol = colTile * 16 + l;

  float scale = 1.0f;
  if (rowscale) scale = 1.0f / fmaxf(rowscale[arow], 1.0f);

  // A 16x4 frag: VGPR0={K0|K2}, VGPR1={K1|K3} -> fold half*2 into the row ptr
  const float* Arow  = A + (size_t)arow * lda + (half << 1);
  const float* Wtile = W + (size_t)colTile * 16;   // tile column base

  v8f acc = {};
  for (int k0 = 0; k0 < K; k0 += KC) {
    const int kc = (K - k0) < KC ? (K - k0) : KC;

    // ---- stage B chunk (kc x 16) into LDS ----
#if USE_TDM
    if (threadIdx.x < 32) {                        // wave 0 drives the TDM
      const unsigned long long ga = (unsigned long long)(const void*)(Wtile + (size_t)k0 * Nc);
      const unsigned lds_addr = (unsigned)(size_t)(void*)ldsB;   // low 32 bits = LDS offset
      uint32x4 g0;
      g0.x = 1u;                                               // count=1, user descriptor
      g0.y = lds_addr;                                         // [63:32]  lds_addr
      g0.z = (unsigned)ga;                                     // [95:64]  global_addr lo
      g0.w = ((unsigned)(ga >> 32) & 0x01FFFFFFu) | (2u << 30); // global_addr hi | type=2
      int32x8 g1;
      g1[0] = (int)(2u << 16);                                 // data_size = 4 bytes
      g1[1] = (int)(16u << 16);                                // tensor_dim0 = 16 (lo bits)
      g1[2] = (int)(((unsigned)kc & 0xFFFFu) << 16);           // tensor_dim1 = kc (lo bits)
      g1[3] = (int)(16u << 16);                                // tile_dim0 = 16
      g1[4] = kc;                                              // tile_dim1 = kc
      g1[5] = Nc;                                              // tensor_dim0_stride = Nc
      g1[6] = 0;
      g1[7] = 0;
      const int32x4 z4 = {0, 0, 0, 0};
#if __clang_major__ >= 23
      const int32x8 z8 = {0, 0, 0, 0, 0, 0, 0, 0};
      __builtin_amdgcn_tensor_load_to_lds(g0, g1, z4, z4, z8, 0);
#else
      __builtin_amdgcn_tensor_load_to_lds(g0, g1, z4, z4, 0);
#endif
      __builtin_amdgcn_s_wait_tensorcnt(0);
    }
#else
    for (int i = threadIdx.x; i < kc * 16; i += 256)
      ldsB[i] = Wtile[(size_t)(k0 + (i >> 4)) * Nc + (i & 15)];
#endif
    __syncthreads();

    // ---- WMMA over the chunk ----
    if (valid) {
      for (int kk = 0; kk < kc; kk += 4) {
        __builtin_prefetch(Arow + k0 + kk + 256, 0, 0);        // global_prefetch_b8
        v2f a, b;
        a.x = Arow[k0 + kk]     * scale;
        a.y = Arow[k0 + kk + 1] * scale;
        const int ks = kk + (half << 1);
        b.x = ldsB[ks * 16 + l];                               // ds_load
        b.y = ldsB[(ks + 1) * 16 + l];
        acc = __builtin_amdgcn_wmma_f32_16x16x4_f32(false, a, false, b,
                                                    (short)0, acc, false, false);
      }
    }
    __syncthreads();
  }

  if (!valid) return;

  const float bv  = bias ? bias[bcol] : 0.0f;
  const int   act = (flags >> 1) & 3;
#pragma unroll
  for (int v = 0; v < 8; ++v) {
    const int row = rowTile * 16 + (half << 3) + v;  // C layout: VGPR v -> M = v (+8 hi half)
    const size_t idx = (size_t)row * ldc + bcol;
    float val = acc[v] + bv;
    if (flags & 1) val += C[idx];
    C[idx] = act_apply(val, act);
  }
}

// ---------------------------------------------------------------------------
// Per-relation edge aggregation: agg[dst] += x[src] for edges of relation r
// (x and agg are L2-resident: 25.6 MB each vs 192 MB L2)
// ---------------------------------------------------------------------------
__global__ void edge_agg_kernel(const float* __restrict__ X,
                                const int* __restrict__ src,
                                const int* __restrict__ dst,
                                const int* __restrict__ et,
                                int r, float* __restrict__ agg, float* __restrict__ cnt)
{
  const size_t total  = (size_t)N_EDGES * 64;
  const size_t stride = (size_t)gridDim.x * blockDim.x;
  for (size_t i = blockIdx.x * (size_t)blockDim.x + threadIdx.x; i < total; i += stride) {
    const size_t e = i >> 6;
    const int    c = (int)(i & 63);
    if (et[e] == r) {
      const int d = dst[e], s = src[e];
      atomicAdd(&agg[(size_t)d * 64 + c], X[(size_t)s * 64 + c]);
      if (c == 0) atomicAdd(&cnt[d], 1.0f);
    }
  }
}

// ---------------------------------------------------------------------------
// Fills
// ---------------------------------------------------------------------------
__global__ void fill_f32(float* __restrict__ p, size_t n, float v) {
  const size_t stride = (size_t)gridDim.x * blockDim.x;
  for (size_t i = blockIdx.x * (size_t)blockDim.x + threadIdx.x; i < n; i += stride) p[i] = v;
}
__global__ void fill_i32(int* __restrict__ p, size_t n, int v) {
  const size_t stride = (size_t)gridDim.x * blockDim.x;
  for (size_t i = blockIdx.x * (size_t)blockDim.x + threadIdx.x; i < n; i += stride) p[i] = v;
}

// ---------------------------------------------------------------------------
// BatchNorm (training-mode batch stats)
// ---------------------------------------------------------------------------
__global__ void col_stats_kernel(const float* __restrict__ X, int M, int Cc, int ld,
                                 float* __restrict__ stats /* [2*Cc] */)
{
  const int t     = blockIdx.x * blockDim.x + threadIdx.x;
  const int total = gridDim.x * blockDim.x;          // multiple of Cc by construction
  const int c       = t % Cc;
  const int r0      = t / Cc;
  const int rstride = total / Cc;
  float s = 0.0f, s2 = 0.0f;
  for (int r = r0; r < M; r += rstride) {
    const float v = X[(size_t)r * ld + c];
    s += v; s2 += v * v;
  }
  atomicAdd(&stats[c], s);
  atomicAdd(&stats[Cc + c], s2);
}

__global__ void bn_finalize_kernel(const float* __restrict__ stats,
                                   const float* __restrict__ g, const float* __restrict__ b,
                                   int Cc, float invM, float* __restrict__ ss /* [2*Cc] */)
{
  const int c = blockIdx.x * blockDim.x + threadIdx.x;
  if (c >= Cc) return;
  const float m   = stats[c] * invM;
  const float var = stats[Cc + c] * invM - m * m;
  const float sc  = g[c] * rsqrtf(var + EPSV);
  ss[c]      = sc;
  ss[Cc + c] = b[c] - m * sc;
}

__global__ void bn_apply_kernel(float* __restrict__ X, const float* __restrict__ ss,
                                int M, int Cc, int ld, int act)
{
  const size_t total  = (size_t)M * Cc;
  const size_t stride = (size_t)gridDim.x * blockDim.x;
  for (size_t i = blockIdx.x * (size_t)blockDim.x + threadIdx.x; i < total; i += stride) {
    const size_t r = i / Cc;
    const int    c = (int)(i - r * Cc);
    const size_t idx = r * ld + c;
    X[idx] = act_apply(X[idx] * ss[c] + ss[Cc + c], act);
  }
}

// ---------------------------------------------------------------------------
// Global mean/max pooling per graph (inputs are post-ReLU => >= 0, so signed-int
// atomicMax with -inf init is exact and preserves -inf for empty graphs)
// ---------------------------------------------------------------------------
__global__ void pool_scatter_kernel(const float* __restrict__ G, const int* __restrict__ batch,
                                    float* __restrict__ sums, int* __restrict__ maxs,
                                    float* __restrict__ gcnt)
{
  const size_t total  = (size_t)N_NODES * HID;
  const size_t stride = (size_t)gridDim.x * blockDim.x;
  for (size_t i = blockIdx.x * (size_t)blockDim.x + threadIdx.x; i < total; i += stride) {
    const size_t node = i >> 6;
    const int    c    = (int)(i & 63);
    const int    b    = batch[node];
    const float  v    = G[i];
    atomicAdd(&sums[(size_t)b * 64 + c], v);
    atomicMax(&maxs[(size_t)b * 64 + c], __float_as_int(v));
    if (c == 0) atomicAdd(&gcnt[b], 1.0f);
  }
}

__global__ void pool_finalize_kernel(const float* __restrict__ sums, const int* __restrict__ maxs,
                                     const float* __restrict__ gcnt, float* __restrict__ gvec)
{
  const int i = blockIdx.x * blockDim.x + threadIdx.x;
  if (i >= BATCH * 128) return;
  const int b = i >> 7, c = i & 127;
  gvec[i] = (c < 64) ? sums[(size_t)b * 64 + c] / fmaxf(gcnt[b], 1.0f)
                     : __int_as_float(maxs[(size_t)b * 64 + (c - 64)]);
}

// ---------------------------------------------------------------------------
// Host-side helpers
// ---------------------------------------------------------------------------
static inline void gemm(hipStream_t s, const float* A, const float* W, const float* bias,
                        const float* rs, float* C, int M, int K, int Nc,
                        int lda, int ldc, int flags)
{
  const int rowTiles  = M / 16;
  const int colTiles  = Nc / 16;
  const int rowGroups = (rowTiles + 7) / 8;
  wmma_gemm_kernel<<<rowGroups * colTiles, 256, 0, s>>>(A, W, bias, rs, C,
                                                        M, K, Nc, lda, ldc, flags);
}

static inline void batchnorm(hipStream_t s, float* X, const float* g, const float* b,
                             int M, int Cc, int ld, int act, float* stats, float* ss)
{
  fill_f32<<<2, 256, 0, s>>>(stats, (size_t)(2 * Cc), 0.0f);
  col_stats_kernel<<<Cc / 2, 256, 0, s>>>(X, M, Cc, ld, stats);   // Cc*128 threads
  bn_finalize_kernel<<<1, 256, 0, s>>>(stats, g, b, Cc, 1.0f / (float)M, ss);
  bn_apply_kernel<<<4096, 256, 0, s>>>(X, ss, M, Cc, ld, act);
}

static void rgcn_layer(hipStream_t s, const float* xin, float* out,
                       const float* W, const float* root, const float* bias,
                       const int* src, const int* dst, const int* et,
                       float* agg, float* cnt)
{
  // out = x @ root + bias
  gemm(s, xin, root, bias, nullptr, out, N_NODES, HID, HID, HID, HID, 0);
  // out += mean-agg_r @ W_r, one relation at a time through one agg buffer
  for (int r = 0; r < RELS; ++r) {
    fill_f32<<<4096, 256, 0, s>>>(agg, (size_t)N_NODES * HID, 0.0f);
    fill_f32<<<512, 256, 0, s>>>(cnt, (size_t)N_NODES, 0.0f);
    edge_agg_kernel<<<8192, 256, 0, s>>>(xin, src, dst, et, r, agg, cnt);
    gemm(s, agg, W + (size_t)r * HID * HID, nullptr, cnt, out,
         N_NODES, HID, HID, HID, HID, /*accumulate*/ 1);
  }
}

extern "C" void kernel_launch(void* const* d_in, const int* in_sizes, int n_in,
                              void* d_out, int out_size, void* d_ws, size_t ws_size,
                              hipStream_t stream)
{
  // --- inputs (setup_inputs dict order) ---
  const float* x          = (const float*)d_in[0];
  const float* ngram_feat = (const float*)d_in[1];
  const int*   eidx       = (const int*)d_in[2];
  const int*   etype      = (const int*)d_in[3];
  const int*   batch      = (const int*)d_in[4];
  const float* conv1_W    = (const float*)d_in[5];
  const float* conv1_root = (const float*)d_in[6];
  const float* conv1_b    = (const float*)d_in[7];
  const float* bn1_g      = (const float*)d_in[8];
  const float* bn1_b      = (const float*)d_in[9];
  const float* conv2_W    = (const float*)d_in[10];
  const float* conv2_root = (const float*)d_in[11];
  const float* conv2_b    = (const float*)d_in[12];
  const float* bn2_g      = (const float*)d_in[13];
  const float* bn2_b      = (const float*)d_in[14];
  const float* gp_W       = (const float*)d_in[15];
  const float* gp_b       = (const float*)d_in[16];
  const float* gpbn_g     = (const float*)d_in[17];
  const float* gpbn_b     = (const float*)d_in[18];
  const float* ng1_W      = (const float*)d_in[19];
  const float* ng1_b      = (const float*)d_in[20];
  const float* ngbn1_g    = (const float*)d_in[21];
  const float* ngbn1_b    = (const float*)d_in[22];
  const float* ng2_W      = (const float*)d_in[23];
  const float* ng2_b      = (const float*)d_in[24];
  const float* ngbn2_g    = (const float*)d_in[25];
  const float* ngbn2_b    = (const float*)d_in[26];
  const float* ng3_W      = (const float*)d_in[27];
  const float* ng3_b      = (const float*)d_in[28];
  const float* enc1_W     = (const float*)d_in[29];
  const float* enc1_b     = (const float*)d_in[30];
  const float* enc2_W     = (const float*)d_in[31];
  const float* enc2_b     = (const float*)d_in[32];
  const float* dec1_W     = (const float*)d_in[33];
  const float* dec1_b     = (const float*)d_in[34];
  const float* dec2_W     = (const float*)d_in[35];
  const float* dec2_b     = (const float*)d_in[36];

  const int* src = eidx;
  const int* dst = eidx + N_EDGES;

  // --- outputs: pred [B,128] then emb [B,16] ---
  float* pred = (float*)d_out;
  float* emb  = pred + (size_t)BATCH * OUTD;

  // --- workspace carve-up (~81 MB) ---
  float* p     = (float*)d_ws;
  float* h1    = p; p += (size_t)N_NODES * HID;
  float* h2    = p; p += (size_t)N_NODES * HID;
  float* agg   = p; p += (size_t)N_NODES * HID;
  float* cnt   = p; p += (size_t)N_NODES;
  float* stats = p; p += 512;
  float* ss    = p; p += 512;
  float* sums  = p; p += (size_t)BATCH * HID;
  int*   maxs  = (int*)p; p += (size_t)BATCH * HID;
  float* gcnt  = p; p += BATCH;
  float* gvec  = p; p += (size_t)BATCH * 128;
  float* n1    = p; p += (size_t)BATCH * 256;
  float* n2    = p; p += (size_t)BATCH * 128;
  float* comb  = p; p += (size_t)BATCH * 128;
  float* e1    = p; p += (size_t)BATCH * 64;
  float* dd1   = p; p += (size_t)BATCH * 64;
  (void)n_in; (void)in_sizes; (void)out_size; (void)ws_size;

  // --- GNN tower ---
  rgcn_layer(stream, x, h1, conv1_W, conv1_root, conv1_b, src, dst, etype, agg, cnt);
  batchnorm(stream, h1, bn1_g, bn1_b, N_NODES, 64, 64, /*relu*/2, stats, ss);
  rgcn_layer(stream, h1, h2, conv2_W, conv2_root, conv2_b, src, dst, etype, agg, cnt);
  batchnorm(stream, h2, bn2_g, bn2_b, N_NODES, 64, 64, /*relu*/2, stats, ss);

  // --- pooling -> g_vec [B,128] ---
  fill_f32<<<256, 256, 0, stream>>>(sums, (size_t)BATCH * 64, 0.0f);
  fill_i32<<<256, 256, 0, stream>>>(maxs, (size_t)BATCH * 64, (int)0xFF800000); // -inf
  fill_f32<<<8, 256, 0, stream>>>(gcnt, (size_t)BATCH, 0.0f);
  pool_scatter_kernel<<<8192, 256, 0, stream>>>(h2, batch, sums, maxs, gcnt);
  pool_finalize_kernel<<<(BATCH * 128) / 256, 256, 0, stream>>>(sums, maxs, gcnt, gvec);

  // --- graph_proj: Linear(128,64) + BN + LeakyReLU -> comb[:, :64] ---
  gemm(stream, gvec, gp_W, gp_b, nullptr, comb, BATCH, 128, 64, 128, 128, 0);
  batchnorm(stream, comb, gpbn_g, gpbn_b, BATCH, 64, 128, /*lrelu*/1, stats, ss);

  // --- ngram tower -> comb[:, 64:128] ---
  gemm(stream, ngram_feat, ng1_W, ng1_b, nullptr, n1, BATCH, NGRAMD, 256, NGRAMD, 256, 0);
  batchnorm(stream, n1, ngbn1_g, ngbn1_b, BATCH, 256, 256, 1, stats, ss);
  gemm(stream, n1, ng2_W, ng2_b, nullptr, n2, BATCH, 256, 128, 256, 128, 0);
  batchnorm(stream, n2, ngbn2_g, ngbn2_b, BATCH, 128, 128, 1, stats, ss);
  gemm(stream, n2, ng3_W, ng3_b, nullptr, comb + 64, BATCH, 128, 64, 128, 128, /*lrelu*/ (1 << 1));

  // --- encoder_final -> emb ---
  gemm(stream, comb, enc1_W, enc1_b, nullptr, e1, BATCH, 128, 64, 128, 64, (1 << 1));
  gemm(stream, e1, enc2_W, enc2_b, nullptr, emb, BATCH, 64, EMBD, 64, EMBD, 0);

  // --- decoder -> pred ---
  gemm(stream, emb, dec1_W, dec1_b, nullptr, dd1, BATCH, EMBD, 64, EMBD, 64, (1 << 1));
  gemm(stream, dd1, dec2_W, dec2_b, nullptr, pred, BATCH, 64, OUTD, 64, OUTD, 0);
}